// VisionAttention_21955872817363
// MI455X (gfx1250) — compile-verified
//
#include <hip/hip_runtime.h>
#include <math.h>

// ---------------------------------------------------------------------------
// VisionAttention for MI455X (gfx1250), all-f32 via V_WMMA_F32_16X16X4_F32.
// ---------------------------------------------------------------------------

typedef float v2f __attribute__((ext_vector_type(2)));
typedef float v8f __attribute__((ext_vector_type(8)));

#define S_LEN 4096
#define NH    16
#define HD    80
#define HID   1280   // NH * HD

#if defined(__HIP_DEVICE_COMPILE__)
#if !__has_builtin(__builtin_amdgcn_wmma_f32_16x16x4_f32)
#error "missing __builtin_amdgcn_wmma_f32_16x16x4_f32 on this toolchain"
#endif
#endif

// D = A(16x4) * B(4x16) + C, fp32 end to end.
__device__ __forceinline__ v8f wmma4(v2f a, v2f b, v8f c) {
  return __builtin_amdgcn_wmma_f32_16x16x4_f32(
      /*neg_a=*/false, a, /*neg_b=*/false, b,
      /*c_mod=*/(short)0, c, /*reuse_a=*/false, /*reuse_b=*/false);
}

// ---------------------------------------------------------------------------
// Kernel 1: qkv = x @ qkv_w.T + qkv_b, scattered into per-head [H][S][D]
// Q/K/V buffers (RoPE applied later).  One wave per 16x16 tile, 4 waves/block.
// ---------------------------------------------------------------------------
__global__ void k_qkv(const float* __restrict__ x, const float* __restrict__ w,
                      const float* __restrict__ bias,
                      float* __restrict__ Qo, float* __restrict__ Ko,
                      float* __restrict__ Vo) {
  const int lane = threadIdx.x & 31;
  const int wv   = threadIdx.x >> 5;
  const int hi   = lane >> 4;      // half-wave: 0 or 1
  const int ln   = lane & 15;
  const int m0   = blockIdx.x * 16;
  const int n0   = (blockIdx.y * 4 + wv) * 16;

  // A fragment: lane ln holds row m0+ln; VGPR0/1 = K-pair (hi selects K0/K1 vs K2/K3)
  const float* ap = x + (size_t)(m0 + ln) * HID + hi * 2;
  // B fragment: out = x @ w^T -> B(k,n) = w[n,k]; lane ln holds col n0+ln
  const float* bp = w + (size_t)(n0 + ln) * HID + hi * 2;

  v8f acc = {};
#pragma unroll 4
  for (int kk = 0; kk < HID; kk += 4) {
    v2f a = *(const v2f*)(ap + kk);
    v2f b = *(const v2f*)(bp + kk);
    acc = wmma4(a, b, acc);
  }

  const float bv = bias[n0 + ln];
  const int which = n0 / HID;            // 0=q, 1=k, 2=v
  const int rem   = n0 - which * HID;
  const int h     = rem / HD;
  const int d0    = rem - h * HD;        // 16-aligned within the head (80 % 16 == 0)
  float* dst = (which == 0) ? Qo : (which == 1) ? Ko : Vo;
  dst += (size_t)h * S_LEN * HD + d0 + ln;
#pragma unroll
  for (int r = 0; r < 8; ++r) {
    const int srow = m0 + r + hi * 8;    // C layout: row = vgpr + 8*half
    dst[(size_t)srow * HD] = acc[r] + bv;
  }
}

// ---------------------------------------------------------------------------
// Kernel 2: in-place RoPE on Q and K.  emb = concat(freqs,freqs), so element d
// and d+40 share cos/sin; each thread owns one (d, d+40) pair.
// ---------------------------------------------------------------------------
__global__ void k_rope(float* __restrict__ Q, float* __restrict__ K,
                       const float* __restrict__ fr) {
  const int idx = blockIdx.x * blockDim.x + threadIdx.x;  // over S*NH*(HD/2)
  if (idx >= S_LEN * NH * (HD / 2)) return;
  const int dp = idx % (HD / 2);
  const int t  = idx / (HD / 2);
  const int h  = t % NH;
  const int s  = t / NH;

  const float f = fr[(size_t)s * (HD / 2) + dp];
  const float c = cosf(f), sn = sinf(f);
  const size_t base = ((size_t)h * S_LEN + s) * HD + dp;

  float q1 = Q[base], q2 = Q[base + HD / 2];
  Q[base]          = q1 * c - q2 * sn;   // rotate_half: [-t2, t1]
  Q[base + HD / 2] = q2 * c + q1 * sn;

  float k1 = K[base], k2 = K[base + HD / 2];
  K[base]          = k1 * c - k2 * sn;
  K[base + HD / 2] = k2 * c + k1 * sn;
}

// ---------------------------------------------------------------------------
// Kernel 3: block-diagonal flash attention.  One wave per (q-tile, head).
// Scores and PV both via f32 WMMA; online softmax with shfl_xor row reduces
// (masks 1/2/4/8 reduce each 16-lane half independently, matching the C
// fragment's row split); P re-layouted C->A through a 1KB LDS tile.
// ---------------------------------------------------------------------------
__global__ void k_attn(const float* __restrict__ Q, const float* __restrict__ K,
                       const float* __restrict__ V, const int* __restrict__ cu,
                       int ncu, float* __restrict__ out) {
  __shared__ __align__(16) float Pld[16 * 16];

  const int lane = threadIdx.x;
  const int hi = lane >> 4, ln = lane & 15;
  const int i = blockIdx.x;   // query tile (16 rows)
  const int h = blockIdx.y;

  int cuv[8];
#pragma unroll
  for (int t = 0; t < 8; ++t) cuv[t] = (t < ncu) ? cu[t] : 0x7fffffff;
  auto segf = [&](int s) {   // searchsorted(cu[1:], s, 'right')
    int g = 0;
#pragma unroll
    for (int t = 1; t < 8; ++t)
      if (t < ncu) g += (cuv[t] <= s) ? 1 : 0;
    return g;
  };

  int qseg[8];
#pragma unroll
  for (int r = 0; r < 8; ++r) qseg[r] = segf(i * 16 + r + hi * 8);
  const int sg_lo = segf(i * 16);
  const int sg_hi = segf(i * 16 + 15);
  const int j0 = cuv[sg_lo] >> 4;
  const int j1 = (cuv[sg_hi + 1] + 15) >> 4;

  // Q A-fragments for all 20 K-steps (D=80), pre-scaled by 1/sqrt(D).
  const float scl = 1.0f / sqrtf((float)HD);
  v2f qa[HD / 4];
  const float* qb = Q + ((size_t)h * S_LEN + i * 16 + ln) * HD + hi * 2;
#pragma unroll
  for (int kk = 0; kk < HD / 4; ++kk) {
    v2f a = *(const v2f*)(qb + kk * 4);
    a.x *= scl; a.y *= scl;
    qa[kk] = a;
  }

  v8f O[5] = {};
  float m[8], l[8];
#pragma unroll
  for (int r = 0; r < 8; ++r) { m[r] = -3.0e38f; l[r] = 0.0f; }

  for (int j = j0; j < j1; ++j) {
    // ---- S = (Q/sqrt(D)) @ K^T for this 16x16 tile ----
    v8f S = {};
    const float* kb = K + ((size_t)h * S_LEN + j * 16 + ln) * HD + hi * 2;
#pragma unroll
    for (int kk = 0; kk < HD / 4; ++kk) {
      v2f b = *(const v2f*)(kb + kk * 4);
      S = wmma4(qa[kk], b, S);
    }

    // ---- segment mask + online softmax ----
    const int tcol = j * 16 + ln;
    const int ksg  = segf(tcol);
    float Sf[8], rmax[8];
#pragma unroll
    for (int r = 0; r < 8; ++r) {
      float s = (ksg == qseg[r]) ? S[r] : -3.0e38f;
      Sf[r] = s; rmax[r] = s;
    }
#pragma unroll
    for (int off = 8; off >= 1; off >>= 1)
#pragma unroll
      for (int r = 0; r < 8; ++r)
        rmax[r] = fmaxf(rmax[r], __shfl_xor(rmax[r], off, 32));

    float p[8], rsum[8];
#pragma unroll
    for (int r = 0; r < 8; ++r) {
      const float mnew  = fmaxf(m[r], rmax[r]);
      const float scale = __expf(m[r] - mnew);
      p[r] = (Sf[r] <= -1.0e38f) ? 0.0f : __expf(Sf[r] - mnew);
      rsum[r] = p[r];
      m[r] = mnew;
      l[r] *= scale;
      // rescale running output rows by `scale`
#pragma unroll
      for (int dt = 0; dt < 5; ++dt) O[dt][r] *= scale;
    }
#pragma unroll
    for (int off = 8; off >= 1; off >>= 1)
#pragma unroll
      for (int r = 0; r < 8; ++r)
        rsum[r] += __shfl_xor(rsum[r], off, 32);
#pragma unroll
    for (int r = 0; r < 8; ++r) l[r] += rsum[r];

    // ---- re-layout P (C fragment) -> A fragment via LDS ----
    __syncthreads();
#pragma unroll
    for (int r = 0; r < 8; ++r) Pld[(r + hi * 8) * 16 + ln] = p[r];
    __syncthreads();
    v2f pa[4];
#pragma unroll
    for (int ks = 0; ks < 4; ++ks)
      pa[ks] = *(const v2f*)(&Pld[ln * 16 + ks * 4 + hi * 2]);

    // ---- O += P @ V_tile (16x16 x 16x80) ----
    const float* vb = V + ((size_t)h * S_LEN + j * 16) * HD + ln;
#pragma unroll
    for (int dt = 0; dt < 5; ++dt) {
#pragma unroll
      for (int ks = 0; ks < 4; ++ks) {
        const int t0 = ks * 4 + hi * 2;
        v2f b;
        b.x = vb[(size_t)t0 * HD + dt * 16];
        b.y = vb[(size_t)(t0 + 1) * HD + dt * 16];
        O[dt] = wmma4(pa[ks], b, O[dt]);
      }
    }
  }

  // ---- normalize and write back as [S, HID] ----
#pragma unroll
  for (int dt = 0; dt < 5; ++dt)
#pragma unroll
    for (int r = 0; r < 8; ++r) {
      const int srow = i * 16 + r + hi * 8;
      out[(size_t)srow * HID + h * HD + dt * 16 + ln] = O[dt][r] / l[r];
    }
}

// ---------------------------------------------------------------------------
// Kernel 4: out = attn_out @ proj_w.T + proj_b
// ---------------------------------------------------------------------------
__global__ void k_proj(const float* __restrict__ A, const float* __restrict__ w,
                       const float* __restrict__ bias, float* __restrict__ out) {
  const int lane = threadIdx.x & 31;
  const int wv   = threadIdx.x >> 5;
  const int hi   = lane >> 4;
  const int ln   = lane & 15;
  const int m0   = blockIdx.x * 16;
  const int n0   = (blockIdx.y * 4 + wv) * 16;

  const float* ap = A + (size_t)(m0 + ln) * HID + hi * 2;
  const float* bp = w + (size_t)(n0 + ln) * HID + hi * 2;

  v8f acc = {};
#pragma unroll 4
  for (int kk = 0; kk < HID; kk += 4) {
    v2f a = *(const v2f*)(ap + kk);
    v2f b = *(const v2f*)(bp + kk);
    acc = wmma4(a, b, acc);
  }

  const float bv = bias[n0 + ln];
#pragma unroll
  for (int r = 0; r < 8; ++r) {
    const int srow = m0 + r + hi * 8;
    out[(size_t)srow * HID + n0 + ln] = acc[r] + bv;
  }
}

// ---------------------------------------------------------------------------
extern "C" void kernel_launch(void* const* d_in, const int* in_sizes, int n_in,
                              void* d_out, int out_size, void* d_ws, size_t ws_size,
                              hipStream_t stream) {
  const float* x      = (const float*)d_in[0];
  const int*   cu     = (const int*)d_in[1];
  const float* rope   = (const float*)d_in[2];
  const float* qkv_w  = (const float*)d_in[3];
  const float* qkv_b  = (const float*)d_in[4];
  const float* proj_w = (const float*)d_in[5];
  const float* proj_b = (const float*)d_in[6];
  float* out = (float*)d_out;

  const size_t per = (size_t)NH * S_LEN * HD;  // 5,242,880 floats = ~21 MB
  float* Q  = (float*)d_ws;
  float* K  = Q + per;
  float* V  = K + per;
  float* AO = V + per;                         // attention output [S, HID]
  const int ncu = in_sizes[1];

  // 1) QKV projection (40.3 GFLOP, f32 WMMA)
  k_qkv<<<dim3(S_LEN / 16, (3 * HID / 16) / 4), 128, 0, stream>>>(
      x, qkv_w, qkv_b, Q, K, V);

  // 2) RoPE on Q, K
  const int nrope = S_LEN * NH * (HD / 2);
  k_rope<<<(nrope + 255) / 256, 256, 0, stream>>>(Q, K, rope);

  // 3) block-diagonal flash attention (21.5 GFLOP)
  k_attn<<<dim3(S_LEN / 16, NH), 32, 0, stream>>>(Q, K, V, cu, ncu, AO);

  // 4) output projection (13.4 GFLOP)
  k_proj<<<dim3(S_LEN / 16, (HID / 16) / 4), 128, 0, stream>>>(
      AO, proj_w, proj_b, out);
}